// GraphSAGE_41832981463453
// MI455X (gfx1250) — compile-verified
//
#include <hip/hip_runtime.h>
#include <hip/hip_bf16.h>

typedef float v2f __attribute__((ext_vector_type(2)));
typedef float v8f __attribute__((ext_vector_type(8)));

#define DEMB 128
#define BN_EPS 1e-5f

// ---------------------------------------------------------------------------
// Zero-fill (graph-capture-safe replacement for hipMemsetAsync)
// ---------------------------------------------------------------------------
__global__ void fill_zero(float* __restrict__ p, long long n) {
    long long i = (long long)blockIdx.x * blockDim.x + threadIdx.x;
    long long stride = (long long)gridDim.x * blockDim.x;
    for (; i < n; i += stride) p[i] = 0.0f;
}

// ---------------------------------------------------------------------------
// Repack 128x128 row-major weight into WMMA f32 16x16x4 B-fragment order:
//   P[(kk*8 + j)*64 + lane*2 + v] = W[(4*kk + 2*(lane>>4) + v)*128 + j*16 + (lane&15)]
// ---------------------------------------------------------------------------
__global__ void repack_w(const float* __restrict__ W, float* __restrict__ P) {
    int t = blockIdx.x * blockDim.x + threadIdx.x;   // 0..16383
    int v    = t & 1;
    int lane = (t >> 1) & 31;
    int j    = (t >> 6) & 7;
    int kk   = t >> 9;
    int col  = j * 16 + (lane & 15);
    int row  = kk * 4 + (lane >> 4) * 2 + v;
    P[t] = W[row * 128 + col];
}

// ---------------------------------------------------------------------------
// Edge scatter-add: one wave per edge; lane handles 4 contiguous channels.
// ---------------------------------------------------------------------------
__global__ void __launch_bounds__(256)
scatter_add(const float* __restrict__ x, const int* __restrict__ src,
            const int* __restrict__ dst, float* __restrict__ agg, int E) {
    int wave = (int)((blockIdx.x * (unsigned)blockDim.x + threadIdx.x) >> 5);
    int lane = threadIdx.x & 31;
    if (wave >= E) return;
    int s = src[wave];
    int d = dst[wave];
    const float4 v = *((const float4*)(x + (long long)s * DEMB) + lane);
    float* a = agg + (long long)d * DEMB + lane * 4;
    atomicAdd(a + 0, v.x);
    atomicAdd(a + 1, v.y);
    atomicAdd(a + 2, v.z);
    atomicAdd(a + 3, v.w);
}

// ---------------------------------------------------------------------------
// Fused dual GEMM via V_WMMA_F32_16X16X4_F32, compile-time specialized:
//   out[N,128] = act( A@Wa + bias [+ X@Wx] )
// One wave computes a 16x128 output strip. Fragment loads are batched per
// k-step so the scheduler can overlap them with the WMMA chain.
// ---------------------------------------------------------------------------
template <bool HAS_X, bool RELU>
__global__ void __launch_bounds__(256)
gemm128_wmma(const float* __restrict__ A,  const float* __restrict__ Wa,
             const float* __restrict__ X,  const float* __restrict__ Wx,
             const float* __restrict__ bias, float* __restrict__ out,
             int ntiles) {
    int wave = (int)((blockIdx.x * (unsigned)blockDim.x + threadIdx.x) >> 5);
    int lane = threadIdx.x & 31;
    if (wave >= ntiles) return;          // wave-uniform: EXEC stays all-ones
    const int row0 = wave * 16;
    const int m    = lane & 15;
    const int kh   = lane >> 4;

    v8f acc[8] = {};

    const float* arow = A + (long long)(row0 + m) * DEMB + 2 * kh;
    const float* xrow = HAS_X ? (X + (long long)(row0 + m) * DEMB + 2 * kh) : nullptr;

#pragma unroll 2
    for (int kk = 0; kk < 32; ++kk) {
        const float* wb  = Wa + kk * 512 + lane * 2;
        const float* wxb = HAS_X ? (Wx + kk * 512 + lane * 2) : nullptr;

        // Batch all fragment loads for this k-step first.
        v2f a = *(const v2f*)(arow + kk * 4);
        v2f ax;
        v2f b[8], bx[8];
#pragma unroll
        for (int j = 0; j < 8; ++j) b[j] = *(const v2f*)(wb + j * 64);
        if (HAS_X) {
            ax = *(const v2f*)(xrow + kk * 4);
#pragma unroll
            for (int j = 0; j < 8; ++j) bx[j] = *(const v2f*)(wxb + j * 64);
        }

        // Then the WMMA chain.
#pragma unroll
        for (int j = 0; j < 8; ++j) {
            acc[j] = __builtin_amdgcn_wmma_f32_16x16x4_f32(
                false, a, false, b[j], (short)0, acc[j], false, false);
            if (HAS_X) {
                acc[j] = __builtin_amdgcn_wmma_f32_16x16x4_f32(
                    false, ax, false, bx[j], (short)0, acc[j], false, false);
            }
        }
    }

    // C/D layout: VGPR r -> M = r + 8*(lane>>4); N = lane&15
    const int coln  = lane & 15;
    const int mhalf = lane >> 4;
#pragma unroll
    for (int j = 0; j < 8; ++j) {
        int col = j * 16 + coln;
        float bv = bias[col];
#pragma unroll
        for (int r = 0; r < 8; ++r) {
            int row = row0 + r + 8 * mhalf;
            float v = acc[j][r] + bv;
            if (RELU) v = fmaxf(v, 0.0f);
            out[(long long)row * DEMB + col] = v;
        }
    }
}

// Scalar tail for rows not covered by 16-row tiles (unused when N%16==0).
__global__ void gemm128_tail(const float* __restrict__ A, const float* __restrict__ Wl,
                             const float* __restrict__ X, const float* __restrict__ Wr,
                             const float* __restrict__ bias, float* __restrict__ out,
                             int row_begin, int nrows, int do_relu) {
    int t   = blockIdx.x * blockDim.x + threadIdx.x;
    int row = row_begin + t / DEMB;
    int col = t % DEMB;
    if (row >= nrows) return;
    float s = bias[col];
    for (int k = 0; k < DEMB; ++k) {
        s += A[(long long)row * DEMB + k] * Wl[k * DEMB + col];
        if (X) s += X[(long long)row * DEMB + k] * Wr[k * DEMB + col];
    }
    if (do_relu) s = fmaxf(s, 0.0f);
    out[(long long)row * DEMB + col] = s;
}

// ---------------------------------------------------------------------------
// BatchNorm: per-channel sum / sumsq, then normalize + affine + ReLU in place
// ---------------------------------------------------------------------------
__global__ void __launch_bounds__(128)
bn_stats(const float* __restrict__ h, float* __restrict__ stats, int n) {
    int c = threadIdx.x;             // 128 channels
    float s = 0.0f, s2 = 0.0f;
    for (int r = blockIdx.x; r < n; r += gridDim.x) {
        float v = h[(long long)r * DEMB + c];
        s += v; s2 += v * v;
    }
    atomicAdd(&stats[c], s);
    atomicAdd(&stats[DEMB + c], s2);
}

__global__ void __launch_bounds__(256)
bn_apply_relu(float* __restrict__ h, const float* __restrict__ stats,
              const float* __restrict__ g, const float* __restrict__ be, int n) {
    long long idx = (long long)blockIdx.x * blockDim.x + threadIdx.x;
    long long tot = (long long)n * DEMB;
    if (idx >= tot) return;
    int c = (int)(idx & (DEMB - 1));
    float inv_n = 1.0f / (float)n;
    float mu  = stats[c] * inv_n;
    float var = stats[DEMB + c] * inv_n - mu * mu;
    float v = (h[idx] - mu) * rsqrtf(var + BN_EPS) * g[c] + be[c];
    h[idx] = fmaxf(v, 0.0f);
}

// ---------------------------------------------------------------------------
// Head: out[n,2] = h[n,128] @ W3[128,2] + b3   — one wave per row, shfl-reduce
// ---------------------------------------------------------------------------
__global__ void __launch_bounds__(256)
head_gemv(const float* __restrict__ h, const float* __restrict__ W3,
          const float* __restrict__ b3, float* __restrict__ out, int n) {
    int wave = (int)((blockIdx.x * (unsigned)blockDim.x + threadIdx.x) >> 5);
    int lane = threadIdx.x & 31;
    if (wave >= n) return;
    const float* hr = h + (long long)wave * DEMB;
    float o0 = 0.0f, o1 = 0.0f;
#pragma unroll
    for (int i = 0; i < 4; ++i) {
        int c = lane + i * 32;
        float v = hr[c];
        o0 += v * W3[c * 2 + 0];
        o1 += v * W3[c * 2 + 1];
    }
#pragma unroll
    for (int off = 16; off > 0; off >>= 1) {
        o0 += __shfl_down(o0, off, 32);
        o1 += __shfl_down(o1, off, 32);
    }
    if (lane == 0) {
        out[(long long)wave * 2 + 0] = o0 + b3[0];
        out[(long long)wave * 2 + 1] = o1 + b3[1];
    }
}

// ---------------------------------------------------------------------------
// Host orchestration
// ---------------------------------------------------------------------------
extern "C" void kernel_launch(void* const* d_in, const int* in_sizes, int n_in,
                              void* d_out, int out_size, void* d_ws, size_t ws_size,
                              hipStream_t stream) {
    const float* x   = (const float*)d_in[0];
    const int*   ei  = (const int*)  d_in[1];
    const float* Wl0 = (const float*)d_in[2];
    const float* bl0 = (const float*)d_in[3];
    const float* Wr0 = (const float*)d_in[4];
    const float* Wl1 = (const float*)d_in[5];
    const float* bl1 = (const float*)d_in[6];
    const float* Wr1 = (const float*)d_in[7];
    const float* g0  = (const float*)d_in[8];
    const float* be0 = (const float*)d_in[9];
    const float* g1  = (const float*)d_in[10];
    const float* be1 = (const float*)d_in[11];
    const float* W1  = (const float*)d_in[12];
    const float* b1  = (const float*)d_in[13];
    const float* W2  = (const float*)d_in[14];
    const float* b2  = (const float*)d_in[15];
    const float* W3  = (const float*)d_in[16];
    const float* b3  = (const float*)d_in[17];
    float* out = (float*)d_out;

    const int N = in_sizes[0] / DEMB;
    const int E = in_sizes[1] / 2;
    const int* src = ei;
    const int* dst = ei + E;

    char* ws = (char*)d_ws;
    const size_t NB = (size_t)N * DEMB * sizeof(float);
    float* bufA = (float*)(ws);               // agg / mlp-h1
    float* bufB = (float*)(ws + NB);          // h0 / mlp-h2
    float* bufC = (float*)(ws + 2 * NB);      // h1
    float* wp   = (float*)(ws + 3 * NB);      // 6 packed 128x128 weights
    float* pWl0 = wp + 0 * 16384;
    float* pWr0 = wp + 1 * 16384;
    float* pWl1 = wp + 2 * 16384;
    float* pWr1 = wp + 3 * 16384;
    float* pW1  = wp + 4 * 16384;
    float* pW2  = wp + 5 * 16384;
    float* stats = wp + 6 * 16384;            // 256 floats: sum | sumsq

    const int ntiles = N / 16;
    const int rem_rows = N - ntiles * 16;
    const dim3 gemm_grid((ntiles + 7) / 8);
    const dim3 blk256(256);
    const long long nelem = (long long)N * DEMB;
    const int elem_blocks = (int)((nelem + 255) / 256);
    const int edge_blocks = (E + 7) / 8;
    const int row_blocks  = (N + 7) / 8;

    // 0) repack weights into WMMA B-fragment order (cheap, deterministic)
    repack_w<<<64, blk256, 0, stream>>>(Wl0, pWl0);
    repack_w<<<64, blk256, 0, stream>>>(Wr0, pWr0);
    repack_w<<<64, blk256, 0, stream>>>(Wl1, pWl1);
    repack_w<<<64, blk256, 0, stream>>>(Wr1, pWr1);
    repack_w<<<64, blk256, 0, stream>>>(W1,  pW1);
    repack_w<<<64, blk256, 0, stream>>>(W2,  pW2);

    // ---- Layer 0 -----------------------------------------------------------
    fill_zero<<<2048, blk256, 0, stream>>>(bufA, nelem);
    scatter_add<<<edge_blocks, blk256, 0, stream>>>(x, src, dst, bufA, E);
    gemm128_wmma<true, false><<<gemm_grid, blk256, 0, stream>>>(
        bufA, pWl0, x, pWr0, bl0, bufB, ntiles);
    if (rem_rows > 0)
        gemm128_tail<<<(rem_rows * DEMB + 255) / 256, blk256, 0, stream>>>(
            bufA, Wl0, x, Wr0, bl0, bufB, ntiles * 16, N, 0);
    fill_zero<<<1, blk256, 0, stream>>>(stats, 256);
    bn_stats<<<1024, dim3(128), 0, stream>>>(bufB, stats, N);
    bn_apply_relu<<<elem_blocks, blk256, 0, stream>>>(bufB, stats, g0, be0, N);

    // ---- Layer 1 -----------------------------------------------------------
    fill_zero<<<2048, blk256, 0, stream>>>(bufA, nelem);
    scatter_add<<<edge_blocks, blk256, 0, stream>>>(bufB, src, dst, bufA, E);
    gemm128_wmma<true, false><<<gemm_grid, blk256, 0, stream>>>(
        bufA, pWl1, bufB, pWr1, bl1, bufC, ntiles);
    if (rem_rows > 0)
        gemm128_tail<<<(rem_rows * DEMB + 255) / 256, blk256, 0, stream>>>(
            bufA, Wl1, bufB, Wr1, bl1, bufC, ntiles * 16, N, 0);
    fill_zero<<<1, blk256, 0, stream>>>(stats, 256);
    bn_stats<<<1024, dim3(128), 0, stream>>>(bufC, stats, N);
    bn_apply_relu<<<elem_blocks, blk256, 0, stream>>>(bufC, stats, g1, be1, N);

    // ---- MLP head ----------------------------------------------------------
    gemm128_wmma<false, true><<<gemm_grid, blk256, 0, stream>>>(
        bufC, pW1, nullptr, nullptr, b1, bufA, ntiles);
    if (rem_rows > 0)
        gemm128_tail<<<(rem_rows * DEMB + 255) / 256, blk256, 0, stream>>>(
            bufC, W1, nullptr, nullptr, b1, bufA, ntiles * 16, N, 1);
    gemm128_wmma<false, true><<<gemm_grid, blk256, 0, stream>>>(
        bufA, pW2, nullptr, nullptr, b2, bufB, ntiles);
    if (rem_rows > 0)
        gemm128_tail<<<(rem_rows * DEMB + 255) / 256, blk256, 0, stream>>>(
            bufA, W2, nullptr, nullptr, b2, bufB, ntiles * 16, N, 1);
    head_gemv<<<row_blocks, blk256, 0, stream>>>(bufB, W3, b3, out, N);
}